// DEFORM_func_55353538511635
// MI455X (gfx1250) — compile-verified
//
#include <hip/hip_runtime.h>
#include <stdint.h>

// Problem constants (match reference)
#define BB   16384
#define NVV  129
#define NE   128
#define RODS 32          // rods per block == one wave32; lane = rod

// TDM staging: buffer0 = vertices 0..63 (192 floats/rod), buffer1 = vertices 64..128 (195 floats/rod)
// TDM LDS padding: 1 dword inserted after every 64 stored dwords -> lds pos = n + (n>>6)
#define B0_STORE (RODS * 192)                       // 6144 floats
#define B0_FOOT  (B0_STORE + (B0_STORE / 64))       // 6240 floats
#define B1_STORE (RODS * 195)                       // 6240 floats
#define B1_FOOT  (B1_STORE + (B1_STORE / 64) + 4)   // 6341 floats (slack)
#define LDS_FLOATS (B0_FOOT + B1_FOOT)

typedef unsigned int u32x4 __attribute__((ext_vector_type(4)));
typedef int          i32x8 __attribute__((ext_vector_type(8)));
typedef int          i32x4 __attribute__((ext_vector_type(4)));

struct F3 { float x, y, z; };

__device__ inline F3 cross3(F3 a, F3 b) {
    return { a.y * b.z - a.z * b.y, a.z * b.x - a.x * b.z, a.x * b.y - a.y * b.x };
}
__device__ inline float dot3(F3 a, F3 b) { return a.x * b.x + a.y * b.y + a.z * b.z; }
__device__ inline F3 sub3(F3 a, F3 b) { return { a.x - b.x, a.y - b.y, a.z - b.z }; }
__device__ inline F3 scl3(F3 a, float s) { return { a.x * s, a.y * s, a.z * s }; }
__device__ inline F3 norm3(F3 a) {
    float n   = sqrtf(dot3(a, a));
    float inv = 1.0f / fmaxf(n, 1e-12f);
    return scl3(a, inv);
}

// LDS reads through the TDM pad mapping: stored dword n lives at n + (n>>6)
__device__ inline float ld_pad(const float* sm, int n) { return sm[n + (n >> 6)]; }

__device__ inline F3 read_v0(const float* sm, int lane, int j) {   // j in [0,63]
    int n = lane * 192 + j * 3;
    return { ld_pad(sm, n), ld_pad(sm, n + 1), ld_pad(sm, n + 2) };
}
__device__ inline F3 read_v1(const float* sm, int lane, int j) {   // j in [64,128]
    int n = lane * 195 + (j - 64) * 3;
    const float* b1 = sm + B0_FOOT;
    return { ld_pad(b1, n), ld_pad(b1, n + 1), ld_pad(b1, n + 2) };
}

// One parallel-transport scan step (edge i >= 1). Also emits kb[i], b_u[i], b_v[i].
__device__ inline void rod_step(F3 vn, F3& vcur, F3& e_prev, float& l_prev, float li,
                                F3& u, F3& v,
                                float* __restrict__ kb_out,
                                float* __restrict__ bu_out,
                                float* __restrict__ bv_out) {
    F3 e = sub3(vn, vcur);
    float denom = l_prev * li + dot3(e_prev, e);
    F3 kb = scl3(cross3(e_prev, e), 2.0f / denom);
    kb_out[0] = kb.x; kb_out[1] = kb.y; kb_out[2] = kb.z;

    float mag    = dot3(kb, kb);
    float inv4   = 1.0f / (4.0f + mag);
    float cosPhi = sqrtf(4.0f * inv4);
    float sinPhi = sqrtf(mag * inv4);
    float kn     = sqrtf(mag);
    F3 axis = scl3(kb, sinPhi / fmaxf(kn, 1e-12f));

    // quat rotate: u + 2*(w*(axis x u) + axis x (axis x u))
    F3 t  = cross3(axis, u);
    F3 ct = cross3(axis, t);
    F3 urot = { u.x + 2.0f * (cosPhi * t.x + ct.x),
                u.y + 2.0f * (cosPhi * t.y + ct.y),
                u.z + 2.0f * (cosPhi * t.z + ct.z) };
    F3 vrot = norm3(cross3(e, urot));

    bool keep = (1.0f - cosPhi) <= 1e-6f;
    u.x = keep ? u.x : urot.x; u.y = keep ? u.y : urot.y; u.z = keep ? u.z : urot.z;
    v.x = keep ? v.x : vrot.x; v.y = keep ? v.y : vrot.y; v.z = keep ? v.z : vrot.z;

    bu_out[0] = u.x; bu_out[1] = u.y; bu_out[2] = u.z;
    bv_out[0] = v.x; bv_out[1] = v.y; bv_out[2] = v.z;

    e_prev = e; l_prev = li; vcur = vn;
}

__global__ __launch_bounds__(RODS) void rod_transport_kernel(
    const float* __restrict__ verts,   // (B, 129, 3)
    const float* __restrict__ initd,   // (B, 3)
    const float* __restrict__ rl,      // (B, 128)
    float* __restrict__ out)           // [b_u | b_v | kb], each (B,128,3)
{
    __shared__ float sm[LDS_FLOATS];

    const int lane = threadIdx.x;
    const int b0   = blockIdx.x * RODS;
    const int b    = b0 + lane;

    // ---- Build TDM descriptors (uniform across the wave) and issue both DMAs ----
    uint32_t lds_base = (uint32_t)(uintptr_t)(void*)sm;          // low 32 bits = LDS offset
    uint64_t g_rod = (uint64_t)(uintptr_t)verts
                   + (uint64_t)b0 * (uint64_t)(NVV * 3 * sizeof(float));
    uint64_t gA = g_rod;                 // vertices 0..63  (192 floats/rod)
    uint64_t gB = g_rod + 192ull * 4ull; // vertices 64..128 (195 floats/rod)

    // Group0: count=1 | lds_addr | global_addr | type=2
    u32x4 g0A = { 1u, lds_base,
                  (uint32_t)gA, (uint32_t)(gA >> 32) | (2u << 30) };
    u32x4 g0B = { 1u, lds_base + (uint32_t)(B0_FOOT * 4),
                  (uint32_t)gB, (uint32_t)(gB >> 32) | (2u << 30) };

    // Group1: data_size=4B, pad_enable=1, pad_interval=64 dwords (enc 5), pad_amount=1 dword (enc 0)
    const int padcfg = (2 << 16) | (1 << 20) | (5 << 22);
    // 2-D tile: tile_dim0 = floats/rod, tile_dim1 = RODS rods, tensor stride0 = 387 floats
    i32x8 g1A = { padcfg, (192 << 16), (RODS << 16), (192 << 16), RODS, 387, 0, 0 };
    i32x8 g1B = { padcfg, (195 << 16), (RODS << 16), (195 << 16), RODS, 387, 0, 0 };
    i32x4 z4  = { 0, 0, 0, 0 };
    i32x8 z8  = { 0, 0, 0, 0, 0, 0, 0, 0 };

    __builtin_amdgcn_tensor_load_to_lds(g0A, g1A, z4, z4, z8, 0);  // TENSORcnt -> 1
    __builtin_amdgcn_tensor_load_to_lds(g0B, g1B, z4, z4, z8, 0);  // TENSORcnt -> 2

    // ---- Per-rod scalar state while DMAs fly ----
    F3 d = { initd[(size_t)b * 3 + 0], initd[(size_t)b * 3 + 1], initd[(size_t)b * 3 + 2] };

    float* bu  = out;
    float* bv  = out + (size_t)BB * NE * 3;
    float* kbo = out + (size_t)2 * BB * NE * 3;
    const size_t ob = (size_t)b * NE * 3;

    // ---- Wait for first tile only; overlap with second DMA ----
    __builtin_amdgcn_s_wait_tensorcnt(1);
    asm volatile("" ::: "memory");

    F3 vcur  = read_v0(sm, lane, 0);
    F3 vnext = read_v0(sm, lane, 1);
    F3 e0 = sub3(vnext, vcur);
    F3 N0 = cross3(e0, d);
    F3 u  = norm3(cross3(N0, e0));
    F3 v  = norm3(cross3(e0, u));

    bu[ob + 0] = u.x; bu[ob + 1] = u.y; bu[ob + 2] = u.z;
    bv[ob + 0] = v.x; bv[ob + 1] = v.y; bv[ob + 2] = v.z;
    kbo[ob + 0] = 0.0f; kbo[ob + 1] = 0.0f; kbo[ob + 2] = 0.0f;

    F3 e_prev   = e0;
    float l_prev = rl[(size_t)b * NE + 0];
    vcur = vnext;

    // Phase A: edges 1..62 read vertices j=i+1 in [2,63] from buffer0
    for (int i = 1; i <= 62; ++i) {
        F3 vn = read_v0(sm, lane, i + 1);
        float li = rl[(size_t)b * NE + i];
        rod_step(vn, vcur, e_prev, l_prev, li, u, v,
                 kbo + ob + (size_t)i * 3, bu + ob + (size_t)i * 3, bv + ob + (size_t)i * 3);
    }

    // Phase B: wait for second tile, edges 63..127 read vertices j in [64,128] from buffer1
    __builtin_amdgcn_s_wait_tensorcnt(0);
    asm volatile("" ::: "memory");

    for (int i = 63; i < NE; ++i) {
        F3 vn = read_v1(sm, lane, i + 1);
        float li = rl[(size_t)b * NE + i];
        rod_step(vn, vcur, e_prev, l_prev, li, u, v,
                 kbo + ob + (size_t)i * 3, bu + ob + (size_t)i * 3, bv + ob + (size_t)i * 3);
    }
}

extern "C" void kernel_launch(void* const* d_in, const int* in_sizes, int n_in,
                              void* d_out, int out_size, void* d_ws, size_t ws_size,
                              hipStream_t stream) {
    const float* verts = (const float*)d_in[0];   // (B,129,3) fp32
    const float* initd = (const float*)d_in[1];   // (B,3)     fp32
    const float* rl    = (const float*)d_in[2];   // (B,128)   fp32
    float* out = (float*)d_out;                   // b_u | b_v | kb, fp32

    dim3 grid(BB / RODS);
    dim3 block(RODS);
    hipLaunchKernelGGL(rod_transport_kernel, grid, block, 0, stream,
                       verts, initd, rl, out);
}